// LatticeCompanderBlockZeroMeanBananaSplit_70274254897548
// MI455X (gfx1250) — compile-verified
//
#include <hip/hip_runtime.h>
#include <hip/hip_bf16.h>
#include <math.h>

// ---------------------------------------------------------------------------
// Sizes
#define B_    4096
#define NBLK_ 12
#define D_    2
#define DY_   2
#define NDIM_ 24
#define NMC_  256
#define DH_   100
#define NTOK_ (B_*NBLK_)          // 49152 token rows for g_a / g_s
#define XHAT_N (NTOK_*D_)         // 98304 floats of x_hat in d_out

typedef __attribute__((ext_vector_type(2))) float v2f;
typedef __attribute__((ext_vector_type(8))) float v8f;

// ---------------------------------------------------------------------------
// device helpers
__device__ __forceinline__ float sp_(float x) {        // softplus, stable
    return fmaxf(x, 0.f) + __logf(1.f + __expf(-fabsf(x)));
}
__device__ __forceinline__ float tanh_(float x) {      // tanh via exp (TRANS op)
    return 1.f - 2.f / (__expf(2.f * x) + 1.f);
}
__device__ __forceinline__ void hexq_(float px, float py, float& qx, float& qy) {
    const float S3  = 1.7320508075688772f;
    const float IS3 = 0.57735026918962576f;
    const float HS  = 0.86602540378443865f;
    float c0x = rintf(px);
    float c0y = rintf(py * IS3) * S3;
    float c1x = rintf(px - 0.5f) + 0.5f;
    float c1y = rintf((py - HS) * IS3) * S3 + HS;
    float d0 = (px - c0x) * (px - c0x) + (py - c0y) * (py - c0y);
    float d1 = (px - c1x) * (px - c1x) + (py - c1y) * (py - c1y);
    bool p0 = (d0 <= d1);
    qx = p0 ? c0x : c1x;
    qy = p0 ? c0y : c1y;
}

// ---------------------------------------------------------------------------
// WMMA f32 GEMM: Y[N,O] = act(X[N,KD] @ W[O,KD]^T + bias)
// One wave per 16-row tile; it owns ALL output columns (OT 16-wide tiles),
// so each A load feeds OT independent back-to-back WMMAs.
// Fully branchless inner loop: K handled at compile time (KD%4==0 -> no
// guards; KD=2 -> v_cndmask select); column overflow handled by clamping
// load addresses (garbage stays in never-stored columns >= O).
// ACT: 0=none 1=leaky_relu(0.01) 2=softplus  (compile-time, like HAS_BIAS)
template<int KD, int OT, int ACT, bool HAS_BIAS>
__global__ __launch_bounds__(32)
void k_gemm16(const float* __restrict__ X, const float* __restrict__ W,
              const float* __restrict__ bias, float* __restrict__ Y,
              int O)
{
    const int lane = threadIdx.x;          // 0..31
    const int hi   = lane >> 4;            // K half-select (VGPR0->K0/K2, VGPR1->K1/K3)
    const int l15  = lane & 15;
    const int row  = blockIdx.x * 16 + l15;

    v8f c[OT];
#pragma unroll
    for (int t = 0; t < OT; ++t) c[t] = (v8f){0.f,0.f,0.f,0.f,0.f,0.f,0.f,0.f};

    const int KSTEPS = (KD + 3) / 4;
#pragma unroll 1
    for (int ks = 0; ks < KSTEPS; ++ks) {
        const int ka = ks * 4 + hi * 2;
        v2f a;
        if (KD % 4 == 0) {
            a.x = X[(size_t)row * KD + ka];
            a.y = X[(size_t)row * KD + ka + 1];
        } else {
            float ax = X[(size_t)row * KD + (ka     < KD ? ka     : 0)];
            float ay = X[(size_t)row * KD + (ka + 1 < KD ? ka + 1 : 0)];
            a.x = (ka     < KD) ? ax : 0.f;
            a.y = (ka + 1 < KD) ? ay : 0.f;
        }
#pragma unroll
        for (int t = 0; t < OT; ++t) {
            const int col = t * 16 + l15;
            const int oc  = (col < O) ? col : (O - 1);   // clamp: garbage cols never stored
            v2f b;
            if (KD % 4 == 0) {
                b.x = W[(size_t)oc * KD + ka];
                b.y = W[(size_t)oc * KD + ka + 1];
            } else {
                float bx = W[(size_t)oc * KD + (ka     < KD ? ka     : 0)];
                float by = W[(size_t)oc * KD + (ka + 1 < KD ? ka + 1 : 0)];
                b.x = (ka     < KD) ? bx : 0.f;
                b.y = (ka + 1 < KD) ? by : 0.f;
            }
            c[t] = __builtin_amdgcn_wmma_f32_16x16x4_f32(
                       /*neg_a=*/false, a, /*neg_b=*/false, b,
                       /*c_mod=*/(short)0, c[t], /*reuse_a=*/false, /*reuse_b=*/false);
        }
    }

#pragma unroll
    for (int t = 0; t < OT; ++t) {
        const int col = t * 16 + l15;
        if (col < O) {
            float bv = 0.f;
            if (HAS_BIAS) bv = bias[col];
#pragma unroll
            for (int r = 0; r < 8; ++r) {
                const int m = r + hi * 8;          // C/D: VGPR r -> M=r / M=r+8
                float v = c[t][r] + bv;
                if (ACT == 1)       v = (v > 0.f) ? v : 0.01f * v;
                else if (ACT == 2)  v = sp_(v);
                Y[(size_t)(blockIdx.x * 16 + m) * O + col] = v;
            }
        }
    }
}

// ---------------------------------------------------------------------------
// ym[c] = 0.05*y_mean[c] + 0.95*mean_b(y1[b,c]),  c in 0..23
__global__ void k_mean(const float* __restrict__ y1, const float* __restrict__ y_mean,
                       float* __restrict__ ym)
{
    __shared__ float red[256];
    const int c = blockIdx.x, tid = threadIdx.x;
    float s = 0.f;
    for (int b = tid; b < B_; b += 256) s += y1[(size_t)b * NDIM_ + c];
    red[tid] = s; __syncthreads();
    for (int st = 128; st > 0; st >>= 1) {
        if (tid < st) red[tid] += red[tid + st];
        __syncthreads();
    }
    if (tid == 0) ym[c] = 0.05f * y_mean[c] + 0.95f * (red[0] * (1.f / B_));
}

// z[i] = y1[i] - ym[i%24]
__global__ void k_subym(const float* __restrict__ y1, const float* __restrict__ ym,
                        float* __restrict__ z, int n)
{
    int i = blockIdx.x * blockDim.x + threadIdx.x;
    if (i < n) z[i] = y1[i] - ym[i % NDIM_];
}
// yh1[i] = ysc[i] + ym[i%24]
__global__ void k_addym(const float* __restrict__ ysc, const float* __restrict__ ym,
                        float* __restrict__ yh1, int n)
{
    int i = blockIdx.x * blockDim.x + threadIdx.x;
    if (i < n) yh1[i] = ysc[i] + ym[i % NDIM_];
}

// yhat = hex_quantize(yc), over npairs 2D pairs
__global__ void k_hexq(const float* __restrict__ yc, float* __restrict__ yhat, int npairs)
{
    int i = blockIdx.x * blockDim.x + threadIdx.x;
    if (i >= npairs) return;
    float px = yc[2 * i], py = yc[2 * i + 1], qx, qy;
    hexq_(px, py, qx, qy);
    yhat[2 * i] = qx; yhat[2 * i + 1] = qy;
}

// u2 = wrap(u @ G) into Voronoi cell; G = kron(I12, [[1,0],[0.5,sqrt3/2]])
__global__ void k_u2(const float* __restrict__ u, float* __restrict__ u2)
{
    int i = blockIdx.x * blockDim.x + threadIdx.x;   // over 256*12 pairs
    if (i >= NMC_ * NBLK_) return;
    int mc = i / NBLK_, j = i % NBLK_;
    float a = u[(size_t)mc * NDIM_ + 2 * j];
    float b = u[(size_t)mc * NDIM_ + 2 * j + 1];
    float px = a + 0.5f * b;
    float py = 0.86602540378443865f * b;
    float qx, qy; hexq_(px, py, qx, qy);
    u2[(size_t)mc * NDIM_ + 2 * j]     = px - qx;
    u2[(size_t)mc * NDIM_ + 2 * j + 1] = py - qy;
}

// ---------------------------------------------------------------------------
// Precompute per-channel constants of the factorized prior:
//   softplus(H*), tanh(a*), copy biases.  Layout (stride 64 per channel):
//   [0:3) sp(H0)  [3:6) b0  [6:9) tanh(a0)
//   per hidden layer L: [9+15L) sp(H)[9], b[3], tanh(a)[3]
//   [54:57) sp(H4)  [57] b4
__global__ void k_params(const float* __restrict__ H0, const float* __restrict__ H1,
                         const float* __restrict__ H2, const float* __restrict__ H3,
                         const float* __restrict__ H4,
                         const float* __restrict__ b0, const float* __restrict__ b1,
                         const float* __restrict__ b2, const float* __restrict__ b3,
                         const float* __restrict__ b4,
                         const float* __restrict__ a0, const float* __restrict__ a1,
                         const float* __restrict__ a2, const float* __restrict__ a3,
                         float* __restrict__ prm)
{
    int c = threadIdx.x;
    if (c >= NDIM_) return;
    float* P = prm + c * 64;
    for (int i = 0; i < 3; ++i) {
        P[i]     = sp_(H0[c * 3 + i]);
        P[3 + i] = b0[c * 3 + i];
        P[6 + i] = tanh_(a0[c * 3 + i]);
    }
    const float* Hs[3] = {H1, H2, H3};
    const float* bs[3] = {b1, b2, b3};
    const float* as[3] = {a1, a2, a3};
    for (int L = 0; L < 3; ++L) {
        float* Q = P + 9 + L * 15;
        for (int j = 0; j < 9; ++j) Q[j] = sp_(Hs[L][c * 9 + j]);
        for (int i = 0; i < 3; ++i) {
            Q[9 + i]  = bs[L][c * 3 + i];
            Q[12 + i] = tanh_(as[L][c * 3 + i]);
        }
    }
    for (int i = 0; i < 3; ++i) P[54 + i] = sp_(H4[c * 3 + i]);
    P[57] = b4[c];
}

// ---------------------------------------------------------------------------
// Monte-Carlo likelihood: one block per batch row, one thread per MC sample.
// Evaluates cdf and its derivative (forward-mode) through the gated net,
// accumulates sum_c log(pdf_c + 1e-9), then block-wide logsumexp.
__global__ __launch_bounds__(256)
void k_mc(const float* __restrict__ yhat, const float* __restrict__ u2,
          const float* __restrict__ prm, float* __restrict__ lik)
{
    __shared__ float sP[NDIM_ * 64];
    __shared__ float sy[NDIM_];
    __shared__ float red[256];
    const int b = blockIdx.x, tid = threadIdx.x;
    for (int i = tid; i < NDIM_ * 64; i += 256) sP[i] = prm[i];
    if (tid < NDIM_) sy[tid] = yhat[(size_t)b * NDIM_ + tid];
    __syncthreads();

    const float* U = u2 + (size_t)tid * NDIM_;
    float logp = 0.f;
#pragma unroll 1
    for (int c = 0; c < NDIM_; ++c) {
        const float* P = sP + c * 64;
        const float x = sy[c] + U[c];
        // layer 0 (1->3) + gate;  d* carries d/dx
        float h0 = P[0] * x + P[3], d0 = P[0];
        float h1 = P[1] * x + P[4], d1 = P[1];
        float h2 = P[2] * x + P[5], d2 = P[2];
        float t;
        t = tanh_(h0); h0 += P[6] * t; d0 *= 1.f + P[6] * (1.f - t * t);
        t = tanh_(h1); h1 += P[7] * t; d1 *= 1.f + P[7] * (1.f - t * t);
        t = tanh_(h2); h2 += P[8] * t; d2 *= 1.f + P[8] * (1.f - t * t);
#pragma unroll
        for (int L = 0; L < 3; ++L) {       // layers 1..3 (3->3) + gates
            const float* Q = P + 9 + L * 15;
            float n0 = Q[0] * h0 + Q[1] * h1 + Q[2] * h2 + Q[9];
            float n1 = Q[3] * h0 + Q[4] * h1 + Q[5] * h2 + Q[10];
            float n2 = Q[6] * h0 + Q[7] * h1 + Q[8] * h2 + Q[11];
            float e0 = Q[0] * d0 + Q[1] * d1 + Q[2] * d2;
            float e1 = Q[3] * d0 + Q[4] * d1 + Q[5] * d2;
            float e2 = Q[6] * d0 + Q[7] * d1 + Q[8] * d2;
            t = tanh_(n0); n0 += Q[12] * t; e0 *= 1.f + Q[12] * (1.f - t * t);
            t = tanh_(n1); n1 += Q[13] * t; e1 *= 1.f + Q[13] * (1.f - t * t);
            t = tanh_(n2); n2 += Q[14] * t; e2 *= 1.f + Q[14] * (1.f - t * t);
            h0 = n0; h1 = n1; h2 = n2; d0 = e0; d1 = e1; d2 = e2;
        }
        const float lg = P[54] * h0 + P[55] * h1 + P[56] * h2 + P[57];
        const float dl = P[54] * d0 + P[55] * d1 + P[56] * d2;
        const float s  = 1.f / (1.f + __expf(-lg));
        const float pdf = s * (1.f - s) * dl;
        logp += __logf(pdf + 1e-9f);
    }

    // block logsumexp over the 256 MC samples
    red[tid] = logp; __syncthreads();
    for (int st = 128; st > 0; st >>= 1) {
        if (tid < st) red[tid] = fmaxf(red[tid], red[tid + st]);
        __syncthreads();
    }
    const float m = red[0]; __syncthreads();
    red[tid] = __expf(logp - m); __syncthreads();
    for (int st = 128; st > 0; st >>= 1) {
        if (tid < st) red[tid] += red[tid + st];
        __syncthreads();
    }
    if (tid == 0) {
        const float LOG_NMC = 5.5451774444795625f;   // log 256
        const float LOG_VOL = -1.7260924865415262f;  // 12*log(sqrt3/2)
        lik[b] = m + __logf(red[0]) - LOG_NMC + LOG_VOL;
    }
}

// ---------------------------------------------------------------------------
extern "C" void kernel_launch(void* const* d_in, const int* in_sizes, int n_in,
                              void* d_out, int out_size, void* d_ws, size_t ws_size,
                              hipStream_t stream)
{
    (void)in_sizes; (void)n_in; (void)out_size; (void)ws_size;
    const float* x      = (const float*)d_in[0];
    const float* y_mean = (const float*)d_in[1];
    const float* u      = (const float*)d_in[2];
    const float* ga_w0  = (const float*)d_in[3];  const float* ga_b0 = (const float*)d_in[4];
    const float* ga_w1  = (const float*)d_in[5];  const float* ga_b1 = (const float*)d_in[6];
    const float* ga_w2  = (const float*)d_in[7];  const float* ga_b2 = (const float*)d_in[8];
    const float* gs_w0  = (const float*)d_in[9];  const float* gs_b0 = (const float*)d_in[10];
    const float* gs_w1  = (const float*)d_in[11]; const float* gs_b1 = (const float*)d_in[12];
    const float* gs_w2  = (const float*)d_in[13]; const float* gs_b2 = (const float*)d_in[14];
    const float* gac_w0 = (const float*)d_in[15];
    const float* gac_w1 = (const float*)d_in[16];
    const float* gac_w2 = (const float*)d_in[17];
    const float* gsc_w0 = (const float*)d_in[18];
    const float* gsc_w1 = (const float*)d_in[19];
    const float* gsc_w2 = (const float*)d_in[20];
    const float* eb_H0  = (const float*)d_in[21];
    const float* eb_H1  = (const float*)d_in[22];
    const float* eb_H2  = (const float*)d_in[23];
    const float* eb_H3  = (const float*)d_in[24];
    const float* eb_H4  = (const float*)d_in[25];
    const float* eb_b0  = (const float*)d_in[26];
    const float* eb_b1  = (const float*)d_in[27];
    const float* eb_b2  = (const float*)d_in[28];
    const float* eb_b3  = (const float*)d_in[29];
    const float* eb_b4  = (const float*)d_in[30];
    const float* eb_a0  = (const float*)d_in[31];
    const float* eb_a1  = (const float*)d_in[32];
    const float* eb_a2  = (const float*)d_in[33];
    const float* eb_a3  = (const float*)d_in[34];

    float* out   = (float*)d_out;
    float* x_hat = out;                 // 98304 floats
    float* lik   = out + XHAT_N;        // 4096 floats

    // workspace carve-up (floats)
    float* ws = (float*)d_ws;
    size_t off = 0;
    float* bufA = ws + off; off += (size_t)NTOK_ * DH_;   // 49152x100
    float* bufB = ws + off; off += (size_t)NTOK_ * DH_;   // 49152x100
    float* y1   = ws + off; off += (size_t)B_ * NDIM_;
    float* ym   = ws + off; off += 32;
    float* z    = ws + off; off += (size_t)B_ * NDIM_;
    float* yc   = ws + off; off += (size_t)B_ * NDIM_;
    float* yhat = ws + off; off += (size_t)B_ * NDIM_;
    float* u2   = ws + off; off += (size_t)NMC_ * NDIM_;
    float* prm  = ws + off; off += (size_t)NDIM_ * 64;
    float* ysc  = ws + off; off += (size_t)B_ * NDIM_;
    float* yh1  = ws + off; off += (size_t)B_ * NDIM_;

    const dim3 blk32(32), blk256(256);

    // ---- g_a : (49152,2) -> 100 -> 100 -> 2 ---------------------------------
    k_gemm16<2,  7,1,true ><<<NTOK_ / 16, blk32, 0, stream>>>(x,    ga_w0, ga_b0, bufA, DH_);
    k_gemm16<100,7,1,true ><<<NTOK_ / 16, blk32, 0, stream>>>(bufA, ga_w1, ga_b1, bufB, DH_);
    k_gemm16<100,1,0,true ><<<NTOK_ / 16, blk32, 0, stream>>>(bufB, ga_w2, ga_b2, y1,   D_);

    // ---- EMA mean, subtract -------------------------------------------------
    k_mean<<<NDIM_, blk256, 0, stream>>>(y1, y_mean, ym);
    k_subym<<<(B_ * NDIM_ + 255) / 256, blk256, 0, stream>>>(y1, ym, z, B_ * NDIM_);

    // ---- g_a_c : (4096,24) -> 100 -> 100 -> 24 (softplus, no bias) ----------
    k_gemm16<24, 7,2,false><<<B_ / 16, blk32, 0, stream>>>(z,    gac_w0, nullptr, bufA, DH_);
    k_gemm16<100,7,2,false><<<B_ / 16, blk32, 0, stream>>>(bufA, gac_w1, nullptr, bufB, DH_);
    k_gemm16<100,2,0,false><<<B_ / 16, blk32, 0, stream>>>(bufB, gac_w2, nullptr, yc,   NDIM_);

    // ---- lattice quantize, MC sample prep -----------------------------------
    k_hexq<<<(B_ * NBLK_ + 255) / 256, blk256, 0, stream>>>(yc, yhat, B_ * NBLK_);
    k_u2<<<(NMC_ * NBLK_ + 255) / 256, blk256, 0, stream>>>(u, u2);
    k_params<<<1, blk32, 0, stream>>>(eb_H0, eb_H1, eb_H2, eb_H3, eb_H4,
                                      eb_b0, eb_b1, eb_b2, eb_b3, eb_b4,
                                      eb_a0, eb_a1, eb_a2, eb_a3, prm);

    // ---- Monte-Carlo likelihood (dominant cost) -----------------------------
    k_mc<<<B_, blk256, 0, stream>>>(yhat, u2, prm, lik);

    // ---- g_s_c : (4096,24) -> 100 -> 100 -> 24, then +ym --------------------
    k_gemm16<24, 7,2,false><<<B_ / 16, blk32, 0, stream>>>(yhat, gsc_w0, nullptr, bufA, DH_);
    k_gemm16<100,7,2,false><<<B_ / 16, blk32, 0, stream>>>(bufA, gsc_w1, nullptr, bufB, DH_);
    k_gemm16<100,2,0,false><<<B_ / 16, blk32, 0, stream>>>(bufB, gsc_w2, nullptr, ysc,  NDIM_);
    k_addym<<<(B_ * NDIM_ + 255) / 256, blk256, 0, stream>>>(ysc, ym, yh1, B_ * NDIM_);

    // ---- g_s : (49152,2) -> 100 -> 100 -> 2  (x_hat straight into d_out) ----
    k_gemm16<2,  7,1,true ><<<NTOK_ / 16, blk32, 0, stream>>>(yh1,  gs_w0, gs_b0, bufA,  DH_);
    k_gemm16<100,7,1,true ><<<NTOK_ / 16, blk32, 0, stream>>>(bufA, gs_w1, gs_b1, bufB,  DH_);
    k_gemm16<100,1,0,true ><<<NTOK_ / 16, blk32, 0, stream>>>(bufB, gs_w2, gs_b2, x_hat, D_);
}